// RealClosedFormContinuous_19756849562236
// MI455X (gfx1250) — compile-verified
//
#include <hip/hip_runtime.h>
#include <math.h>

typedef __attribute__((ext_vector_type(2))) float v2f;
typedef __attribute__((ext_vector_type(8))) float v8f;

#define B_SZ 32
#define T_SZ 128
#define IN_SZ 128
#define H_SZ 256
#define OUT_SZ 128
#define LDS_PAD 258   // 256 + 2: de-conflicts stride-K LDS reads of the A fragment

__device__ __forceinline__ v8f wmma_f32(v2f a, v2f b, v8f c) {
  // V_WMMA_F32_16X16X4_F32 : D = A(16x4) x B(4x16) + C(16x16), fp32 throughout
  return __builtin_amdgcn_wmma_f32_16x16x4_f32(
      /*neg_a=*/false, a, /*neg_b=*/false, b,
      /*c_mod=*/(short)0, c, /*reuse_a=*/false, /*reuse_b=*/false);
}

__device__ __forceinline__ float sigmoidf_(float x) {
  return 1.0f / (1.0f + __expf(-x));
}

// ---------------------------------------------------------------------------
// dst[c*R + r] = src[r*C + c]   (one-time weight transposes, reads coalesced)
// ---------------------------------------------------------------------------
__global__ void transpose_k(const float* __restrict__ src, float* __restrict__ dst,
                            int R, int C) {
  const int idx = blockIdx.x * 256 + threadIdx.x;
  if (idx < R * C) {
    const int r = idx / C, c = idx % C;
    dst[(size_t)c * R + r] = src[idx];
  }
}

// A_effT[k*H + n] = A_eff[n][k] = W_A[n][k] / tau[k]
__global__ void aeffT_kernel(const float* __restrict__ W_A,
                             const float* __restrict__ tau,
                             float* __restrict__ A_effT) {
  const int idx = blockIdx.x * 256 + threadIdx.x;
  if (idx < H_SZ * H_SZ) {
    const int k = idx >> 8, n = idx & (H_SZ - 1);
    A_effT[idx] = W_A[(size_t)n * H_SZ + k] / tau[k];
  }
}

// ---------------------------------------------------------------------------
// NN WMMA GEMM: Out[M,N] = X[M,K] * Wkn[K,N] + bias.
// Block = 256 threads (8 waves) covering MT M-tiles x 8 N-tiles.
// Each wave keeps MT accumulators and reuses one B fragment across MT
// M-tiles -> B (weight) L2 traffic cut by MT. A tiles staged in LDS
// (coalesced, padded rows). N/K/MT are template constants so all strides
// lower to immediate offsets.
// ---------------------------------------------------------------------------
template <int N, int K, int MT>
__global__ void wmma_gemm_nn(const float* __restrict__ X,
                             const float* __restrict__ Wkn,
                             const float* __restrict__ bias,
                             float* __restrict__ Out) {
  constexpr int KP = K + 2;
  __shared__ float a_lds[16 * MT * KP];                 // <= 33.3 KB
  const int lane = threadIdx.x & 31;
  const int wave = threadIdx.x >> 5;
  constexpr int NBLK = N / 128;
  const int mb = blockIdx.x / NBLK;
  const int ntb = (blockIdx.x % NBLK) * 8;
  const int row0 = mb * (16 * MT);

  // Stage A tiles [16*MT x K] coalesced
  for (int idx = threadIdx.x; idx < 16 * MT * K; idx += 256) {
    const int r = idx / K, c = idx % K;                 // K = power of two
    a_lds[r * KP + c] = X[(size_t)(row0 + r) * K + c];
  }
  __syncthreads();

  const int n = (ntb + wave) * 16 + (lane & 15);
  const int khalf = (lane >> 4) * 2;
  const int arow = (lane & 15) * KP;
  const float* bp = Wkn + (size_t)khalf * N + n;

  v8f acc[MT];
#pragma unroll
  for (int j = 0; j < MT; ++j) acc[j] = (v8f){};

  for (int k0 = 0; k0 < K; k0 += 4) {
    v2f b;
    b.x = bp[(size_t)k0 * N];
    b.y = bp[(size_t)(k0 + 1) * N];
#pragma unroll
    for (int j = 0; j < MT; ++j) {
      v2f a;
      a.x = a_lds[j * (16 * KP) + arow + k0 + khalf];
      a.y = a_lds[j * (16 * KP) + arow + k0 + khalf + 1];
      acc[j] = wmma_f32(a, b, acc[j]);
    }
  }
  const float bv = bias ? bias[n] : 0.0f;
#pragma unroll
  for (int j = 0; j < MT; ++j) {
#pragma unroll
    for (int i = 0; i < 8; ++i) {
      const int mm = row0 + j * 16 + (lane >> 4) * 8 + i;  // C/D row layout
      Out[(size_t)mm * N + n] = acc[j][i] + bv;
    }
  }
}

// ---------------------------------------------------------------------------
// LSTM scan. 2 blocks x 512 threads (16 waves); block owns 16 batch rows in
// LDS; cell state c in registers across all 128 steps. Wave w owns hidden
// slice [16w,16w+16) and its i/f/g/o gate tiles -> gate fusion is
// register-local. W_hhT is [K=H, N=4H] so B loads are coalesced b32.
// ---------------------------------------------------------------------------
__global__ void lstm_scan(const float* __restrict__ xg,    // [B,T,4H] = x@W_ih^T + b_ih
                          const float* __restrict__ W_hhT, // [H,4H]
                          const float* __restrict__ b_hh,  // [4H]
                          float* __restrict__ bb_out) {    // [B,T,H]
  __shared__ float h_lds[16 * LDS_PAD];                    // 16.5 KB
  const int tid = threadIdx.x;
  const int wave = tid >> 5, lane = tid & 31;
  const int bbase = blockIdx.x * 16;
  const int n = wave * 16 + (lane & 15);                   // hidden column
  const int khalf = (lane >> 4) * 2;
  const int mrow = (lane >> 4) * 8;
  const int arow = (lane & 15) * LDS_PAD;

  float c[8];
#pragma unroll
  for (int i = 0; i < 8; ++i) c[i] = 0.0f;
  for (int idx = tid; idx < 16 * LDS_PAD; idx += blockDim.x) h_lds[idx] = 0.0f;
  __syncthreads();

  const float bhh0 = b_hh[0 * H_SZ + n];
  const float bhh1 = b_hh[1 * H_SZ + n];
  const float bhh2 = b_hh[2 * H_SZ + n];
  const float bhh3 = b_hh[3 * H_SZ + n];

  for (int t = 0; t < T_SZ; ++t) {
    v8f acc0 = {}, acc1 = {}, acc2 = {}, acc3 = {};
    for (int k0 = 0; k0 < H_SZ; k0 += 4) {
      v2f a;
      a.x = h_lds[arow + k0 + khalf];
      a.y = h_lds[arow + k0 + khalf + 1];
      const size_t r0 = (size_t)(k0 + khalf) * (4 * H_SZ);
      const size_t r1 = r0 + 4 * H_SZ;
      v2f b;
      b.x = W_hhT[r0 + 0 * H_SZ + n]; b.y = W_hhT[r1 + 0 * H_SZ + n];
      acc0 = wmma_f32(a, b, acc0);
      b.x = W_hhT[r0 + 1 * H_SZ + n]; b.y = W_hhT[r1 + 1 * H_SZ + n];
      acc1 = wmma_f32(a, b, acc1);
      b.x = W_hhT[r0 + 2 * H_SZ + n]; b.y = W_hhT[r1 + 2 * H_SZ + n];
      acc2 = wmma_f32(a, b, acc2);
      b.x = W_hhT[r0 + 3 * H_SZ + n]; b.y = W_hhT[r1 + 3 * H_SZ + n];
      acc3 = wmma_f32(a, b, acc3);
    }
    float hnew[8];
#pragma unroll
    for (int i = 0; i < 8; ++i) {
      const int m = mrow + i;                              // local batch row
      const size_t base = ((size_t)(bbase + m) * T_SZ + t) * 4 * H_SZ;
      const float iv = acc0[i] + xg[base + 0 * H_SZ + n] + bhh0;
      const float fv = acc1[i] + xg[base + 1 * H_SZ + n] + bhh1;
      const float gv = acc2[i] + xg[base + 2 * H_SZ + n] + bhh2;
      const float ov = acc3[i] + xg[base + 3 * H_SZ + n] + bhh3;
      c[i] = sigmoidf_(fv) * c[i] + sigmoidf_(iv) * tanhf(gv);
      const float hv = sigmoidf_(ov) * tanhf(c[i]);
      hnew[i] = hv;
      bb_out[((size_t)(bbase + m) * T_SZ + t) * H_SZ + n] = hv;
    }
    __syncthreads();                                       // all GEMM reads of h done
#pragma unroll
    for (int i = 0; i < 8; ++i) h_lds[(mrow + i) * LDS_PAD + n] = hnew[i];
    __syncthreads();
  }
}

// ---------------------------------------------------------------------------
// CfC scan using exp(At)h = h + ts*(A_eff h) + 0.5*ts^2*(A2 h), A2 = A_eff^2.
// Batches are independent -> 2 blocks x 512 threads, block owns 16 batch
// rows. Wave w computes the SAME (0,w) 16x16 tile of all three GEMMs
// (U1 = h A_eff^T, U2 = h A2^T, U3 = f_h W_A^T) so the pointwise combine is
// register-local. All B operands pre-transposed to [K,N] -> coalesced b32.
// ---------------------------------------------------------------------------
__global__ void cfc_scan(const float* __restrict__ Bx,     // [B,T,H]
                         const float* __restrict__ bb,     // [B,T,H]
                         const float* __restrict__ ts,     // [B,T]
                         const float* __restrict__ A_effT, // [H,H]
                         const float* __restrict__ A2T,    // [H,H]
                         const float* __restrict__ W_AT,   // [H,H]
                         const float* __restrict__ tau,    // [H]
                         const float* __restrict__ sigma,  // [H]
                         float* __restrict__ h_hist) {     // [B,T,H]
  __shared__ float h_lds[16 * LDS_PAD];                    // 16.5 KB
  __shared__ float fh_lds[16 * LDS_PAD];                   // 16.5 KB
  const int tid = threadIdx.x;
  const int wave = tid >> 5, lane = tid & 31;
  const int bbase = blockIdx.x * 16;
  const int n = wave * 16 + (lane & 15);
  const int khalf = (lane >> 4) * 2;
  const int arow = (lane & 15) * LDS_PAD;

  for (int idx = tid; idx < 16 * LDS_PAD; idx += blockDim.x) h_lds[idx] = 0.0f;
  __syncthreads();

  const float inv_tau = 1.0f / tau[n];

  for (int t = 0; t < T_SZ; ++t) {
    // f_h = tanh(h / sigma)
    for (int idx = tid; idx < 16 * H_SZ; idx += blockDim.x) {
      const int r = idx >> 8, col = idx & (H_SZ - 1);
      fh_lds[r * LDS_PAD + col] = tanhf(h_lds[r * LDS_PAD + col] / sigma[col]);
    }
    __syncthreads();

    v8f u1 = {}, u2 = {}, u3 = {};
    for (int k0 = 0; k0 < H_SZ; k0 += 4) {
      v2f ah, af;
      ah.x = h_lds[arow + k0 + khalf];
      ah.y = h_lds[arow + k0 + khalf + 1];
      af.x = fh_lds[arow + k0 + khalf];
      af.y = fh_lds[arow + k0 + khalf + 1];
      const size_t r0 = (size_t)(k0 + khalf) * H_SZ + n;
      const size_t r1 = r0 + H_SZ;
      v2f b;
      b.x = A_effT[r0]; b.y = A_effT[r1];
      u1 = wmma_f32(ah, b, u1);
      b.x = A2T[r0];    b.y = A2T[r1];
      u2 = wmma_f32(ah, b, u2);
      b.x = W_AT[r0];   b.y = W_AT[r1];
      u3 = wmma_f32(af, b, u3);
    }
    __syncthreads();                                       // GEMM reads of h/f_h done

    float hnew[8];
#pragma unroll
    for (int i = 0; i < 8; ++i) {
      const int m = (lane >> 4) * 8 + i;                   // local batch row (C layout)
      const int gb = bbase + m;
      const float tsv = ts[(size_t)gb * T_SZ + t];
      const float hold = h_lds[m * LDS_PAD + n];
      const float bxv = Bx[((size_t)gb * T_SZ + t) * H_SZ + n];
      const float bbv = bb[((size_t)gb * T_SZ + t) * H_SZ + n];
      float hn = hold + tsv * u1[i] + 0.5f * tsv * tsv * u2[i];
      hn += tsv * (bxv + u3[i]) * inv_tau;
      hn += bbv;
      hnew[i] = hn;
      h_hist[((size_t)gb * T_SZ + t) * H_SZ + n] = hn;
    }
#pragma unroll
    for (int i = 0; i < 8; ++i) {
      const int m = (lane >> 4) * 8 + i;
      h_lds[m * LDS_PAD + n] = hnew[i];
    }
    __syncthreads();
  }
}

// ---------------------------------------------------------------------------
extern "C" void kernel_launch(void* const* d_in, const int* in_sizes, int n_in,
                              void* d_out, int out_size, void* d_ws, size_t ws_size,
                              hipStream_t stream) {
  const float* x    = (const float*)d_in[0];   // [B,T,IN]
  const float* tsp  = (const float*)d_in[1];   // [B,T]
  const float* tau  = (const float*)d_in[2];   // [H]
  const float* sig  = (const float*)d_in[3];   // [H]
  const float* W_A  = (const float*)d_in[4];   // [H,H]
  const float* W_B  = (const float*)d_in[5];   // [H,IN]
  const float* b_B  = (const float*)d_in[6];   // [H]
  const float* W_C  = (const float*)d_in[7];   // [OUT,H]
  const float* b_C  = (const float*)d_in[8];   // [OUT]
  const float* W_ih = (const float*)d_in[9];   // [4H,IN]
  const float* W_hh = (const float*)d_in[10];  // [4H,H]
  const float* b_ih = (const float*)d_in[11];  // [4H]
  const float* b_hh = (const float*)d_in[12];  // [4H]
  float* out = (float*)d_out;

  const int BT = B_SZ * T_SZ;                  // 4096 rows

  float* ws     = (float*)d_ws;
  float* xg     = ws;                            // BT*4H = 4,194,304 f
  float* Bx     = xg + (size_t)BT * 4 * H_SZ;    // BT*H
  float* bbuf   = Bx + (size_t)BT * H_SZ;        // BT*H
  float* hhist  = bbuf + (size_t)BT * H_SZ;      // BT*H
  float* W_ihT  = hhist + (size_t)BT * H_SZ;     // 128*1024
  float* W_BT   = W_ihT + 4 * H_SZ * IN_SZ;      // 128*256
  float* W_CT   = W_BT + H_SZ * IN_SZ;           // 256*128
  float* W_hhT  = W_CT + H_SZ * OUT_SZ;          // 256*1024
  float* AeffT  = W_hhT + 4 * H_SZ * H_SZ;       // 256*256
  float* A2T    = AeffT + H_SZ * H_SZ;           // 256*256
  float* W_AT   = A2T + H_SZ * H_SZ;             // 256*256

  // --- one-time weight transposes to [K,N] layout (coalesced GEMM B loads) ---
  transpose_k<<<(4 * H_SZ * IN_SZ + 255) / 256, 256, 0, stream>>>(W_ih, W_ihT, 4 * H_SZ, IN_SZ);
  transpose_k<<<(H_SZ * IN_SZ + 255) / 256, 256, 0, stream>>>(W_B, W_BT, H_SZ, IN_SZ);
  transpose_k<<<(OUT_SZ * H_SZ + 255) / 256, 256, 0, stream>>>(W_C, W_CT, OUT_SZ, H_SZ);
  transpose_k<<<(4 * H_SZ * H_SZ + 255) / 256, 256, 0, stream>>>(W_hh, W_hhT, 4 * H_SZ, H_SZ);
  transpose_k<<<(H_SZ * H_SZ + 255) / 256, 256, 0, stream>>>(W_A, W_AT, H_SZ, H_SZ);
  aeffT_kernel<<<(H_SZ * H_SZ + 255) / 256, 256, 0, stream>>>(W_A, tau, AeffT);

  // A2T = (A_eff^2)^T = A_effT @ A_effT     [256 x 256], K=256, MT=2
  wmma_gemm_nn<H_SZ, H_SZ, 2><<<(H_SZ / 32) * (H_SZ / 128), 256, 0, stream>>>(
      AeffT, AeffT, nullptr, A2T);

  // xg = x @ W_ih^T + b_ih   [4096 x 1024], K=128, MT=4
  wmma_gemm_nn<4 * H_SZ, IN_SZ, 4><<<(BT / 64) * (4 * H_SZ / 128), 256, 0, stream>>>(
      x, W_ihT, b_ih, xg);
  // Bx = x @ W_B^T + b_B     [4096 x 256], K=128, MT=4
  wmma_gemm_nn<H_SZ, IN_SZ, 4><<<(BT / 64) * (H_SZ / 128), 256, 0, stream>>>(
      x, W_BT, b_B, Bx);

  // LSTM scan -> backbone_out
  lstm_scan<<<2, 512, 0, stream>>>(xg, W_hhT, b_hh, bbuf);
  // CfC scan -> h_hist
  cfc_scan<<<2, 512, 0, stream>>>(Bx, bbuf, tsp, AeffT, A2T, W_AT, tau, sig, hhist);

  // y = h_hist @ W_C^T + b_C [4096 x 128], K=256, MT=2
  wmma_gemm_nn<OUT_SZ, H_SZ, 2><<<(BT / 32) * (OUT_SZ / 128), 256, 0, stream>>>(
      hhist, W_CT, b_C, out);
}